// GAT_11914239279615
// MI455X (gfx1250) — compile-verified
//
#include <hip/hip_runtime.h>

#define BATCH 16
#define NN    1024
#define F_IN  128
#define F_OUT 64
#define LRELU_ALPHA 0.2f
#define MASK_VAL   -1000000000000.0f

typedef __attribute__((ext_vector_type(2))) float v2f;
typedef __attribute__((ext_vector_type(4))) float v4f;
typedef __attribute__((ext_vector_type(8))) float v8f;
typedef __attribute__((ext_vector_type(4))) int   v4i;

// XOR swizzle for the score panel in LDS: keeps b64/b128 accesses contiguous
// (mask is a multiple of 4 elements) while spreading the WMMA A-operand
// column reads (16 rows, stride 1024 floats) across all 64 banks.
#define SWZ(row, j) ((j) ^ (((row) & 15) << 2))

// ---------------------------------------------------------------------------
// Kernel 1: h[b,n,o] = sum_f inp[b,n,f] * W[f,o]   (f32 WMMA 16x16x4)
// grid = B * N/16 blocks, 128 threads (4 waves), wave w owns cols 16w..16w+15
// ---------------------------------------------------------------------------
__global__ __launch_bounds__(128) void gat_h_kernel(
    const float* __restrict__ inp, const float* __restrict__ W,
    float* __restrict__ h) {
  const int b    = blockIdx.x >> 6;          // / (N/16)
  const int row0 = (blockIdx.x & 63) << 4;
  const int wave = threadIdx.x >> 5;
  const int lane = threadIdx.x & 31;
  const int half = lane >> 4;
  const int l15  = lane & 15;
  const int col  = (wave << 4) + l15;

  const float* arow = inp + ((size_t)(b * NN + row0 + l15) * F_IN);

  v8f c = {};
#pragma unroll
  for (int k = 0; k < F_IN; k += 4) {
    // A 16x4: lane<16 holds K=k,k+1 of its row; lane>=16 holds K=k+2,k+3
    v2f a = *(const v2f*)(arow + k + 2 * half);
    // B 4x16: vgpr0 = W[k+2*half][col], vgpr1 = W[k+2*half+1][col]
    v2f bb;
    bb.x = W[(k + 2 * half) * F_OUT + col];
    bb.y = W[(k + 2 * half + 1) * F_OUT + col];
    c = __builtin_amdgcn_wmma_f32_16x16x4_f32(false, a, false, bb,
                                              (short)0, c, false, false);
  }
#pragma unroll
  for (int g = 0; g < 8; ++g) {
    const int r = row0 + g + 8 * half;      // C layout: vgpr g = rows g, g+8
    h[(size_t)(b * NN + r) * F_OUT + col] = c[g];
  }
}

// ---------------------------------------------------------------------------
// Kernel 2: s1[r] = h[r,:] . a[0:64],  s2[r] = h[r,:] . a[64:128]
// one thread per row (B*N rows)
// ---------------------------------------------------------------------------
__global__ __launch_bounds__(256) void gat_proj_kernel(
    const float* __restrict__ h, const float* __restrict__ a,
    float* __restrict__ s1, float* __restrict__ s2) {
  const int r = blockIdx.x * 256 + threadIdx.x;
  const v4f* hr = (const v4f*)(h + (size_t)r * F_OUT);
  const v4f* a1 = (const v4f*)a;
  const v4f* a2 = (const v4f*)(a + F_OUT);
  float acc1 = 0.f, acc2 = 0.f;
#pragma unroll
  for (int o = 0; o < F_OUT / 4; ++o) {
    v4f hv = hr[o], x = a1[o], y = a2[o];
    acc1 += hv.x * x.x + hv.y * x.y + hv.z * x.z + hv.w * x.w;
    acc2 += hv.x * y.x + hv.y * y.y + hv.z * y.z + hv.w * y.w;
  }
  s1[r] = acc1;
  s2[r] = acc2;
}

// ---------------------------------------------------------------------------
// Kernel 3: fused mask + softmax + (attn @ h) + ELU
// grid = B * N/16 blocks, 128 threads (4 waves)
// LDS: 16 x 1024 f32 score panel (exactly 64 KB), XOR-swizzled
// ---------------------------------------------------------------------------
__global__ __launch_bounds__(128) void gat_attn_kernel(
    const int* __restrict__ adj, const float* __restrict__ h,
    const float* __restrict__ s1, const float* __restrict__ s2,
    float* __restrict__ out) {
  __shared__ float sc[16 * NN];

  const int b    = blockIdx.x >> 6;
  const int row0 = (blockIdx.x & 63) << 4;
  const int t    = threadIdx.x;

  // ---- softmax phase mapping: 8 threads per query row ----
  const int row  = t >> 3;       // 0..15
  const int sub  = t & 7;        // 0..7
  const float s1v = s1[b * NN + row0 + row];
  const int*   adj_row = adj + (size_t)(b * NN + row0 + row) * NN;
  const float* s2b     = s2 + b * NN;
  float* scr = sc + row * NN;

  // pass 1: e = leaky_relu(s1+s2), mask by adj (streamed NT), track row max
  float mx = -3.0e38f;
  for (int it = 0; it < NN / 32; ++it) {
    const int j = it * 32 + sub * 4;
    __builtin_prefetch(adj_row + j + 256, 0, 0);   // global_prefetch_b8
    v4i av  = __builtin_nontemporal_load((const v4i*)(adj_row + j));
    v4f s2v = *(const v4f*)(s2b + j);
    float e0 = s1v + s2v.x; e0 = e0 > 0.f ? e0 : LRELU_ALPHA * e0;
    float e1 = s1v + s2v.y; e1 = e1 > 0.f ? e1 : LRELU_ALPHA * e1;
    float e2 = s1v + s2v.z; e2 = e2 > 0.f ? e2 : LRELU_ALPHA * e2;
    float e3 = s1v + s2v.w; e3 = e3 > 0.f ? e3 : LRELU_ALPHA * e3;
    v4f v;
    v.x = av.x > 0 ? e0 : MASK_VAL;
    v.y = av.y > 0 ? e1 : MASK_VAL;
    v.z = av.z > 0 ? e2 : MASK_VAL;
    v.w = av.w > 0 ? e3 : MASK_VAL;
    *(v4f*)(scr + SWZ(row, j)) = v;               // swizzled b128 store
    mx = fmaxf(mx, fmaxf(fmaxf(v.x, v.y), fmaxf(v.z, v.w)));
  }
#pragma unroll
  for (int m = 4; m >= 1; m >>= 1) mx = fmaxf(mx, __shfl_xor(mx, m, 8));

  // pass 2a: p = exp(s - mx), accumulate row sum
  float l = 0.f;
  for (int it = 0; it < NN / 32; ++it) {
    const int j = it * 32 + sub * 4;
    float* p = scr + SWZ(row, j);
    v4f v = *(v4f*)p;
    v.x = __expf(v.x - mx);
    v.y = __expf(v.y - mx);
    v.z = __expf(v.z - mx);
    v.w = __expf(v.w - mx);
    *(v4f*)p = v;
    l += v.x + v.y + v.z + v.w;
  }
#pragma unroll
  for (int m = 4; m >= 1; m >>= 1) l += __shfl_xor(l, m, 8);
  const float inv_l = 1.0f / l;

  // pass 2b: normalize in place (attn row now final in LDS)
  for (int it = 0; it < NN / 32; ++it) {
    const int j = it * 32 + sub * 4;
    float* p = scr + SWZ(row, j);
    v4f v = *(v4f*)p;
    v.x *= inv_l; v.y *= inv_l; v.z *= inv_l; v.w *= inv_l;
    *(v4f*)p = v;
  }
  __syncthreads();

  // ---- WMMA phase: out[16 x 64] = attn[16 x 1024] @ h[1024 x 64] ----
  const int wave = t >> 5;
  const int lane = t & 31;
  const int half = lane >> 4;
  const int l15  = lane & 15;
  const int col  = (wave << 4) + l15;
  const float* hb = h + (size_t)b * NN * F_OUT;

  v8f c = {};
#pragma unroll 8
  for (int kk = 0; kk < NN; kk += 4) {
    const int k = kk + 2 * half;
    // A 16x4 from LDS: row = l15, columns k,k+1 (swizzle-safe b64 pair)
    v2f a = *(const v2f*)(sc + l15 * NN + SWZ(l15, k));
    // B 4x16 from global h (L2-resident, reused by 64 blocks per batch)
    v2f bb;
    bb.x = hb[k * F_OUT + col];
    bb.y = hb[(k + 1) * F_OUT + col];
    c = __builtin_amdgcn_wmma_f32_16x16x4_f32(false, a, false, bb,
                                              (short)0, c, false, false);
  }

#pragma unroll
  for (int g = 0; g < 8; ++g) {
    const int rc = g + 8 * half;
    float x = c[g];
    x = x > 0.f ? x : (__expf(x) - 1.0f);          // ELU (alpha=1)
    out[(size_t)(b * NN + row0 + rc) * F_OUT + col] = x;
  }
}

// ---------------------------------------------------------------------------
extern "C" void kernel_launch(void* const* d_in, const int* in_sizes, int n_in,
                              void* d_out, int out_size, void* d_ws, size_t ws_size,
                              hipStream_t stream) {
  const float* inp = (const float*)d_in[0];   // (16,1024,128) f32
  const int*   adj = (const int*)d_in[1];     // (16,1024,1024) i32
  const float* W   = (const float*)d_in[2];   // (128,64) f32
  const float* a   = (const float*)d_in[3];   // (128,1) f32
  float* out = (float*)d_out;                 // (16,1024,64) f32

  float* ws = (float*)d_ws;
  float* h  = ws;                               // 16*1024*64 = 1 M floats
  float* s1 = ws + (size_t)BATCH * NN * F_OUT;  // 16*1024
  float* s2 = s1 + (size_t)BATCH * NN;          // 16*1024

  const dim3 blk128(128), blk256(256);
  const dim3 tiles(BATCH * (NN / 16));          // 1024 blocks

  gat_h_kernel<<<tiles, blk128, 0, stream>>>(inp, W, h);
  gat_proj_kernel<<<dim3((BATCH * NN) / 256), blk256, 0, stream>>>(h, a, s1, s2);
  gat_attn_kernel<<<tiles, blk128, 0, stream>>>(adj, h, s1, s2, out);
}